// concat_Net_88880053223545
// MI455X (gfx1250) — compile-verified
//
#include <hip/hip_runtime.h>

typedef float v2f __attribute__((ext_vector_type(2)));
typedef float v8f __attribute__((ext_vector_type(8)));

#define NN 50000   // multiple of 16 -> no M-edge tiles in the GEMMs
#define NE 1600000
#define NG 128

// ---------------- zero ----------------
__global__ void zero_f32(float* __restrict__ p, int n) {
    int i = blockIdx.x * blockDim.x + threadIdx.x;
    if (i < n) p[i] = 0.0f;
}

// ---------------- degree ----------------
__global__ void degree_kernel(const int* __restrict__ dst, float* __restrict__ deg, int E) {
    int e = blockIdx.x * blockDim.x + threadIdx.x;
    if (e < E) atomicAdd(&deg[dst[e]], 1.0f);
}

// ---------------- fp32 WMMA GEMM: C[M,Nc] = A[M,K] @ B[K,Nc], row-major ----------------
// One wave per 16x16 output tile; K stepped by 4 via v_wmma_f32_16x16x4_f32.
// Requirements: M % 16 == 0, K % 4 == 0. Nc arbitrary (cols clamped on load,
// guarded on store -- garbage columns are computed but never written).
__global__ void gemm_wmma_f32(const float* __restrict__ A, const float* __restrict__ B,
                              float* __restrict__ C, int K, int Nc) {
    const int lane = threadIdx.x;        // 0..31
    const int half = lane >> 4;          // 0 or 1
    const int lid  = lane & 15;
    const int row    = blockIdx.x * 16 + lid;
    const int colRaw = blockIdx.y * 16 + lid;
    const int col    = colRaw < Nc ? colRaw : (Nc - 1);   // branchless clamp

    // Lane-private streaming pointers (all 64-bit math hoisted out of the loop).
    const float* pa = A + row * K + half * 2;     // 8B-aligned: K%4==0, half*2 even
    const float* pb = B + half * 2 * Nc + col;

    v8f acc = {};
    const int steps = K >> 2;
    for (int s = 0; s < steps; ++s) {
        v2f a = *(const v2f*)pa;                  // one b64 load: A[row][ka], A[row][ka+1]
        v2f b;
        b.x = pb[0];                              // B[ka][col]
        b.y = pb[Nc];                             // B[ka+1][col]
        acc = __builtin_amdgcn_wmma_f32_16x16x4_f32(
            false, a, false, b, (short)0, acc, false, false);
        pa += 4;
        pb += 4 * Nc;
    }

    // C/D layout: VGPR r -> M = tile*16 + half*8 + r, N = col. Rows always valid.
    if (colRaw < Nc) {
        float* pc = C + (blockIdx.x * 16 + half * 8) * Nc + colRaw;
#pragma unroll
        for (int r = 0; r < 8; ++r) pc[r * Nc] = acc[r];
    }
}

// ---------------- edge scatter-add: S[dst] += X[src], D % 4 == 0 ----------------
__global__ void scatter_add4(const float* __restrict__ X, const int* __restrict__ src,
                             const int* __restrict__ dst, float* __restrict__ S,
                             int E, int D) {
    const int D4 = D >> 2;
    long gid = (long)blockIdx.x * blockDim.x + threadIdx.x;
    if (gid >= (long)E * D4) return;
    int e = (int)(gid / D4);
    int c = (int)(gid % D4) * 4;
    int s = src[e], d = dst[e];
    const float4 v = *(const float4*)(X + s * D + c);   // 16B-aligned (D*4 % 16 == 0)
    float* p = S + d * D + c;
    atomicAdd(p + 0, v.x);
    atomicAdd(p + 1, v.y);
    atomicAdd(p + 2, v.z);
    atomicAdd(p + 3, v.w);
}

// ---------------- h = relu((deg>0 ? S/deg : X) + b), in-place into S ----------------
__global__ void finalize_layer(float* __restrict__ S, const float* __restrict__ X,
                               const float* __restrict__ deg, const float* __restrict__ bias,
                               int N, int D) {
    long gid = (long)blockIdx.x * blockDim.x + threadIdx.x;
    if (gid >= (long)N * D) return;
    int n = (int)(gid / D);
    int d = (int)(gid % D);
    float dg = deg[n];
    float agg = (dg > 0.0f) ? (S[gid] / dg) : X[gid];
    S[gid] = fmaxf(agg + bias[d], 0.0f);
}

// ---------------- per-graph sum + count ----------------
__global__ void pool_kernel(const float* __restrict__ h, const int* __restrict__ graph_id,
                            float* __restrict__ g_sum, float* __restrict__ g_cnt, int N) {
    long gid = (long)blockIdx.x * blockDim.x + threadIdx.x;
    if (gid >= (long)N * 20) return;
    int n = (int)(gid / 20);
    int d = (int)(gid % 20);
    int g = graph_id[n];
    atomicAdd(&g_sum[g * 20 + d], h[gid]);
    if (d == 0) atomicAdd(&g_cnt[g], 1.0f);
}

// ---------------- final tiny MLP: one thread per graph ----------------
__global__ void mlp_kernel(const float* __restrict__ g_sum, const float* __restrict__ g_cnt,
                           const float* __restrict__ self_feat,
                           const float* __restrict__ Wf1, const float* __restrict__ bf1,
                           const float* __restrict__ Wf2, const float* __restrict__ bf2,
                           float* __restrict__ out) {
    int g = threadIdx.x;
    if (g >= NG) return;
    float fused[36];
    float cnt = fmaxf(g_cnt[g], 1.0f);
#pragma unroll
    for (int d = 0; d < 20; ++d) fused[d] = g_sum[g * 20 + d] / cnt;
#pragma unroll
    for (int d = 0; d < 16; ++d) fused[20 + d] = self_feat[g * 16 + d];
    float o = bf2[0];
#pragma unroll
    for (int j = 0; j < 10; ++j) {
        float s = bf1[j];
#pragma unroll
        for (int i = 0; i < 36; ++i) s += fused[i] * Wf1[i * 10 + j];
        o += fmaxf(s, 0.0f) * Wf2[j];
    }
    out[g] = o;
}

extern "C" void kernel_launch(void* const* d_in, const int* in_sizes, int n_in,
                              void* d_out, int out_size, void* d_ws, size_t ws_size,
                              hipStream_t stream) {
    const float* feat      = (const float*)d_in[0];   // [NN,128]
    const float* self_feat = (const float*)d_in[1];   // [NG,16]
    const int*   src       = (const int*)d_in[2];     // [NE]
    const int*   dst       = (const int*)d_in[3];     // [NE]
    const int*   graph_id  = (const int*)d_in[4];     // [NN]
    const float* W1        = (const float*)d_in[5];   // [128,100]
    const float* b1        = (const float*)d_in[6];   // [100]
    const float* W2        = (const float*)d_in[7];   // [100,20]
    const float* b2        = (const float*)d_in[8];   // [20]
    const float* Wf1       = (const float*)d_in[9];   // [36,10]
    const float* bf1       = (const float*)d_in[10];  // [10]
    const float* Wf2       = (const float*)d_in[11];  // [10,1]
    const float* bf2       = (const float*)d_in[12];  // [1]
    float* out = (float*)d_out;

    // Workspace layout (floats). Regions needing zeroing form a contiguous prefix.
    float* ws    = (float*)d_ws;
    float* deg   = ws;                               // NN
    float* S1    = deg + NN;                         // NN*100
    float* S2    = S1 + (size_t)NN * 100;            // NN*20
    float* g_sum = S2 + (size_t)NN * 20;             // NG*20
    float* g_cnt = g_sum + (size_t)NG * 20;          // NG
    float* X1    = g_cnt + NG;                       // NN*100
    float* X2    = X1 + (size_t)NN * 100;            // NN*20

    const long zeroCount = (long)NN + (long)NN * 100 + (long)NN * 20 + NG * 20 + NG;
    zero_f32<<<(int)((zeroCount + 255) / 256), 256, 0, stream>>>(ws, (int)zeroCount);

    degree_kernel<<<(NE + 255) / 256, 256, 0, stream>>>(dst, deg, NE);

    // Layer 1: transform first (X1 = feat @ W1), then aggregate 100-dim messages
    // (linearity: segment_sum(h[src]) @ W == segment_sum((h@W)[src])).
    dim3 g1(NN / 16, (100 + 15) / 16);
    gemm_wmma_f32<<<g1, 32, 0, stream>>>(feat, W1, X1, 128, 100);
    {
        long tasks = (long)NE * 25;  // 100/4 float4 chunks per edge
        scatter_add4<<<(int)((tasks + 255) / 256), 256, 0, stream>>>(X1, src, dst, S1, NE, 100);
    }
    {
        long tasks = (long)NN * 100;
        finalize_layer<<<(int)((tasks + 255) / 256), 256, 0, stream>>>(S1, X1, deg, b1, NN, 100);
    }

    // Layer 2: X2 = h1 @ W2 (K=100), aggregate 20-dim messages (5x less edge traffic).
    dim3 g2(NN / 16, (20 + 15) / 16);
    gemm_wmma_f32<<<g2, 32, 0, stream>>>(S1, W2, X2, 100, 20);
    {
        long tasks = (long)NE * 5;   // 20/4 float4 chunks per edge
        scatter_add4<<<(int)((tasks + 255) / 256), 256, 0, stream>>>(X2, src, dst, S2, NE, 20);
    }
    {
        long tasks = (long)NN * 20;
        finalize_layer<<<(int)((tasks + 255) / 256), 256, 0, stream>>>(S2, X2, deg, b2, NN, 20);
    }

    // Graph mean pooling + tiny MLP head.
    {
        long tasks = (long)NN * 20;
        pool_kernel<<<(int)((tasks + 255) / 256), 256, 0, stream>>>(S2, graph_id, g_sum, g_cnt, NN);
    }
    mlp_kernel<<<1, 128, 0, stream>>>(g_sum, g_cnt, self_feat, Wf1, bf1, Wf2, bf2, out);
}